// GATClassifier_1176821039335
// MI455X (gfx1250) — compile-verified
//
#include <hip/hip_runtime.h>
#include <hip/hip_bf16.h>

// GCN classifier for MI455X (gfx1250, wave32).
// Pipeline (all on `stream`, graph-capture safe):
//   1. degrees (f32 atomics) -> D_out^-1/2, D_in^-1/2
//   2. pack W1/W2 into WMMA-fragment order (zero-padded K)
//   3. GEMM (fp32 WMMA 16x16x4) with fused out_isqrt row scale -> bufX
//   4. zero bufA; SPMM scatter (wave/edge, global_atomic_add_f32) -> bufA
//   5. fused in_isqrt * x + b, ReLU (in place)
//   6. layer 2 same (ping-pong buffers)
//   7. per-graph mean pool (atomics), tiny FC head -> d_out [1000 x 18]

typedef float v2f __attribute__((ext_vector_type(2)));
typedef float v8f __attribute__((ext_vector_type(8)));

static constexpr int N_NODES = 100000;
static constexpr int N_EDGES = 1600000;
static constexpr int N_GRAPHS = 1000;
static constexpr int F = 256;          // hidden width
static constexpr int K1 = 137, K1P = 140;

// ---------------- utility kernels ----------------

__global__ void zero_f(float* __restrict__ p, int n) {
    int i = blockIdx.x * blockDim.x + threadIdx.x;
    if (i < n) p[i] = 0.0f;
}

__global__ void zero_f4(float4* __restrict__ p, int n4) {
    int i = blockIdx.x * blockDim.x + threadIdx.x;
    if (i < n4) p[i] = make_float4(0.f, 0.f, 0.f, 0.f);
}

__global__ void degree_kernel(const int* __restrict__ src, const int* __restrict__ dst,
                              float* __restrict__ outdeg, float* __restrict__ indeg, int nE) {
    int e = blockIdx.x * blockDim.x + threadIdx.x;
    if (e < nE) {
        atomicAdd(outdeg + src[e], 1.0f);
        atomicAdd(indeg + dst[e], 1.0f);
    }
}

__global__ void isqrt_kernel(float* __restrict__ p, int n) {
    int i = blockIdx.x * blockDim.x + threadIdx.x;
    if (i < n) p[i] = rsqrtf(fmaxf(p[i], 1.0f));   // clip(deg,1)^-0.5
}

// Pack row-major W[K x 256] into WMMA B-fragment order:
//   Wp[(kt*16 + nt)*32 + lane] = { W[k0][col], W[k0+1][col] }
//   k0 = 4*kt + 2*(lane>>4), col = nt*16 + (lane&15); zero-pad k >= K.
__global__ void pack_w_kernel(const float* __restrict__ W, float2* __restrict__ Wp,
                              int K, int Kpad) {
    const int idx = blockIdx.x * blockDim.x + threadIdx.x;   // (Kpad/4)*512 pairs
    const int total = (Kpad >> 2) * 512;
    if (idx >= total) return;
    const int kt   = idx >> 9;
    const int r    = idx & 511;
    const int nt   = r >> 5;
    const int lane = r & 31;
    const int half = lane >> 4, lm = lane & 15;
    const int k0   = (kt << 2) + (half << 1);
    const int col  = (nt << 4) + lm;
    const float x = (k0 < K)     ? W[k0 * F + col]       : 0.0f;
    const float y = (k0 + 1 < K) ? W[(k0 + 1) * F + col] : 0.0f;
    Wp[idx] = make_float2(x, y);
}

// ---------------- fp32 WMMA GEMM: C[N,256] = (A[N,K] @ W[K,256]) * rowscale ----------------
// block = 128 threads (4 waves); block strip = 32 rows; wave w -> N-tiles [4w, 4w+4)
// each wave: 2 M-tiles x 4 N-tiles, K-step 4 via V_WMMA_F32_16X16X4_F32.
// B comes pre-packed in fragment order -> one coalesced global_load_b64 per fragment.

__launch_bounds__(128)
__global__ void gemm_rowscale_wmma(const float* __restrict__ A, int lda, int K, int Kpad,
                                   const float2* __restrict__ Bpack,
                                   const float* __restrict__ rowscale,
                                   float* __restrict__ C) {
    __shared__ float As[32 * 260];                 // stride 260 dwords: conflict-free
    const int tid = threadIdx.x;
    const int rows0 = blockIdx.x * 32;

    // cooperative A-strip load with zero padding to Kpad
    for (int r = 0; r < 32; ++r) {
        for (int c = tid; c < Kpad; c += 128) {
            float v = (c < K) ? A[(size_t)(rows0 + r) * lda + c] : 0.0f;
            As[r * 260 + c] = v;
        }
    }
    __syncthreads();

    const int w    = tid >> 5;     // wave id 0..3
    const int lane = tid & 31;
    const int half = lane >> 4;    // ISA layout: lanes 16-31 carry K+2 / M+8
    const int lm   = lane & 15;

    v8f acc[2][4] = {};

    const int nk = Kpad >> 2;
    for (int kt = 0; kt < nk; ++kt) {
        const int k0 = (kt << 2) + (half << 1);
        // A frags: element (M = lm [+16 for tile 1], K = k0, k0+1)
        v2f a0, a1;
        a0.x = As[lm * 260 + k0];
        a0.y = As[lm * 260 + k0 + 1];
        a1.x = As[(lm + 16) * 260 + k0];
        a1.y = As[(lm + 16) * 260 + k0 + 1];
        const float2* Bp = Bpack + (size_t)kt * 512 + (w << 7) + lane;  // + nt*32
#pragma unroll
        for (int j = 0; j < 4; ++j) {
            const float2 bb = Bp[j << 5];
            v2f b; b.x = bb.x; b.y = bb.y;
            acc[0][j] = __builtin_amdgcn_wmma_f32_16x16x4_f32(
                false, a0, false, b, (short)0, acc[0][j], false, false);
            acc[1][j] = __builtin_amdgcn_wmma_f32_16x16x4_f32(
                false, a1, false, b, (short)0, acc[1][j], false, false);
        }
    }

    // epilogue: C/D layout VGPR i -> M = i + 8*half, N = lm; fuse row scale
#pragma unroll
    for (int m = 0; m < 2; ++m) {
#pragma unroll
        for (int i = 0; i < 8; ++i) {
            const int row = rows0 + m * 16 + i + 8 * half;
            const float s = rowscale[row];
#pragma unroll
            for (int j = 0; j < 4; ++j) {
                const int ncol = (((w << 2) + j) << 4) + lm;
                C[(size_t)row * F + ncol] = acc[m][j][i] * s;
            }
        }
    }
}

// ---------------- SPMM scatter: agg[dst[e]] += x[src[e]]  (one wave per edge) ----------------

__global__ void spmm_scatter(const float* __restrict__ x, const int* __restrict__ src,
                             const int* __restrict__ dst, float* __restrict__ agg, int nE) {
    const int gid = blockIdx.x * blockDim.x + threadIdx.x;
    const int e = gid >> 5;
    if (e >= nE) return;
    const int lane = gid & 31;
    const int s = src[e];
    const int d = dst[e];
    const float4* xs = (const float4*)(x + (size_t)s * F);
    float* ad = agg + (size_t)d * F + lane * 8;
    const float4 v0 = xs[lane * 2];
    const float4 v1 = xs[lane * 2 + 1];
    atomicAdd(ad + 0, v0.x); atomicAdd(ad + 1, v0.y);
    atomicAdd(ad + 2, v0.z); atomicAdd(ad + 3, v0.w);
    atomicAdd(ad + 4, v1.x); atomicAdd(ad + 5, v1.y);
    atomicAdd(ad + 6, v1.z); atomicAdd(ad + 7, v1.w);
}

// ---------------- fused a = relu(a * in_isqrt[row] + bias[col]) ----------------

__global__ void scale_bias_relu(float* __restrict__ a, const float* __restrict__ in_isqrt,
                                const float* __restrict__ bias) {
    const size_t idx = (size_t)blockIdx.x * blockDim.x + threadIdx.x;  // N_NODES*F exact
    const int row = (int)(idx >> 8);
    const int col = (int)(idx & 255);
    const float v = a[idx] * in_isqrt[row] + bias[col];
    a[idx] = fmaxf(v, 0.0f);
}

// ---------------- per-graph mean pool (one wave per node) ----------------

__global__ void pool_kernel(const float* __restrict__ h2, const int* __restrict__ graph_ids,
                            float* __restrict__ sums, float* __restrict__ counts, int nN) {
    const int gid = blockIdx.x * blockDim.x + threadIdx.x;
    const int n = gid >> 5;
    if (n >= nN) return;
    const int lane = gid & 31;
    const int g = graph_ids[n];
    const float4* hp = (const float4*)(h2 + (size_t)n * F);
    float* sp = sums + (size_t)g * F + lane * 8;
    const float4 v0 = hp[lane * 2];
    const float4 v1 = hp[lane * 2 + 1];
    atomicAdd(sp + 0, v0.x); atomicAdd(sp + 1, v0.y);
    atomicAdd(sp + 2, v0.z); atomicAdd(sp + 3, v0.w);
    atomicAdd(sp + 4, v1.x); atomicAdd(sp + 5, v1.y);
    atomicAdd(sp + 6, v1.z); atomicAdd(sp + 7, v1.w);
    if (lane == 0) atomicAdd(counts + g, 1.0f);
}

// ---------------- head: out[g,:] = (sums[g]/max(count,1)) @ Wfc + bfc ----------------

__global__ void final_fc(const float* __restrict__ sums, const float* __restrict__ counts,
                         const float* __restrict__ Wfc, const float* __restrict__ bfc,
                         float* __restrict__ out) {
    const int g = blockIdx.x;
    const int k = threadIdx.x;         // 32 lanes, 18 active
    const float inv = 1.0f / fmaxf(counts[g], 1.0f);
    if (k >= 18) return;
    float accv = bfc[k];
    const float* sg = sums + (size_t)g * F;
    for (int i = 0; i < F; ++i) accv += (sg[i] * inv) * Wfc[i * 18 + k];
    out[g * 18 + k] = accv;
}

// ---------------- launcher ----------------

extern "C" void kernel_launch(void* const* d_in, const int* in_sizes, int n_in,
                              void* d_out, int out_size, void* d_ws, size_t ws_size,
                              hipStream_t stream) {
    (void)in_sizes; (void)n_in; (void)out_size; (void)ws_size;

    const float* h   = (const float*)d_in[0];
    const float* W1  = (const float*)d_in[1];
    const float* b1  = (const float*)d_in[2];
    const float* W2  = (const float*)d_in[3];
    const float* b2  = (const float*)d_in[4];
    const float* Wfc = (const float*)d_in[5];
    const float* bfc = (const float*)d_in[6];
    const int*   src = (const int*)d_in[7];
    const int*   dst = (const int*)d_in[8];
    const int*   gid = (const int*)d_in[9];
    float* out = (float*)d_out;

    // workspace layout (floats)
    float* ws = (float*)d_ws;
    const size_t O_ISQ = 0;                                  // [0,N): out_isqrt, [N,2N): in_isqrt
    const size_t O_WP1 = O_ISQ + 2 * (size_t)N_NODES;        // (K1P/4)*512*2 = 35840
    const size_t O_WP2 = O_WP1 + (size_t)(K1P / 4) * 1024;   // (256/4)*512*2 = 65536
    const size_t O_X   = O_WP2 + (size_t)(F / 4) * 1024;     // N * F  (301376 % 4 == 0 -> 16B aligned)
    const size_t O_A   = O_X + (size_t)N_NODES * F;          // N * F
    const size_t O_SUM = O_A + (size_t)N_NODES * F;          // G * F
    const size_t O_CNT = O_SUM + (size_t)N_GRAPHS * F;       // G

    float* out_isqrt = ws + O_ISQ;
    float* in_isqrt  = ws + O_ISQ + N_NODES;
    float2* Wp1      = (float2*)(ws + O_WP1);
    float2* Wp2      = (float2*)(ws + O_WP2);
    float* bufX      = ws + O_X;
    float* bufA      = ws + O_A;
    float* sums      = ws + O_SUM;
    float* counts    = ws + O_CNT;

    const int nf = N_NODES * F;          // 25.6M
    const int nf4 = nf / 4;

    // 1. degrees -> isqrt
    zero_f<<<(2 * N_NODES + 255) / 256, 256, 0, stream>>>(out_isqrt, 2 * N_NODES);
    degree_kernel<<<(N_EDGES + 255) / 256, 256, 0, stream>>>(src, dst, out_isqrt, in_isqrt, N_EDGES);
    isqrt_kernel<<<(2 * N_NODES + 255) / 256, 256, 0, stream>>>(out_isqrt, 2 * N_NODES);

    // 2. pack weights into WMMA fragment order (zero-padded K)
    pack_w_kernel<<<((K1P / 4) * 512 + 255) / 256, 256, 0, stream>>>(W1, Wp1, K1, K1P);
    pack_w_kernel<<<((F / 4) * 512 + 255) / 256, 256, 0, stream>>>(W2, Wp2, F, F);

    // ---- layer 1 ----
    gemm_rowscale_wmma<<<N_NODES / 32, 128, 0, stream>>>(h, K1, K1, K1P, Wp1, out_isqrt, bufX);
    zero_f4<<<(nf4 + 255) / 256, 256, 0, stream>>>((float4*)bufA, nf4);
    spmm_scatter<<<N_EDGES / 8, 256, 0, stream>>>(bufX, src, dst, bufA, N_EDGES);
    scale_bias_relu<<<N_NODES, 256, 0, stream>>>(bufA, in_isqrt, b1);

    // ---- layer 2 ----
    gemm_rowscale_wmma<<<N_NODES / 32, 128, 0, stream>>>(bufA, F, F, F, Wp2, out_isqrt, bufX);
    zero_f4<<<(nf4 + 255) / 256, 256, 0, stream>>>((float4*)bufA, nf4);
    spmm_scatter<<<N_EDGES / 8, 256, 0, stream>>>(bufX, src, dst, bufA, N_EDGES);
    scale_bias_relu<<<N_NODES, 256, 0, stream>>>(bufA, in_isqrt, b2);

    // ---- pool + head ----
    zero_f<<<(N_GRAPHS * F + N_GRAPHS + 255) / 256, 256, 0, stream>>>(sums, N_GRAPHS * F + N_GRAPHS);
    pool_kernel<<<(N_NODES * 32 + 255) / 256, 256, 0, stream>>>(bufA, gid, sums, counts, N_NODES);
    final_fc<<<N_GRAPHS, 32, 0, stream>>>(sums, counts, Wfc, bfc, out);
}